// Loss_XCL_40742059770681
// MI455X (gfx1250) — compile-verified
//
#include <hip/hip_runtime.h>
#include <hip/hip_bf16.h>
#include <math.h>

// ---------------------------------------------------------------------------
// Fused InfoNCE (both directions) for B=8192, DIM=256, T=0.07 on gfx1250.
//   loss = [ sum_i log(rowSum_i) + sum_j log(colSum_j) - 2*posSum ] / B
// sim = (Q_n R_n^T)/T is never materialized: each 128x128 tile is produced by
// V_WMMA_F32_16X16X4_F32 and immediately folded into the LSE accumulators.
// |sim| <= 14.3 so exp never overflows f32 -> single pass, no max subtraction.
// LDS staging is double-buffered via CDNA5 GLOBAL_LOAD_ASYNC_TO_LDS (ASYNCcnt),
// overlapping the next chunk's DMA with the current chunk's WMMA stream.
// ---------------------------------------------------------------------------

typedef __attribute__((ext_vector_type(2))) float v2f;
typedef __attribute__((ext_vector_type(8))) float v8f;
typedef __attribute__((ext_vector_type(4))) int   v4i;

#define B_TOTAL   8192
#define DIM       256
#define HALF_B    4096
#define INV_T     (1.0f / 0.07f)
#define LDST      36      // LDS row stride (floats): 16B-aligned rows, conflict-free

#define AS1 __attribute__((address_space(1)))
#define AS3 __attribute__((address_space(3)))

#if defined(__has_builtin)
#if __has_builtin(__builtin_amdgcn_global_load_async_to_lds_b128)
#define USE_ASYNC 1
#endif
#endif
#ifndef USE_ASYNC
#define USE_ASYNC 0
#endif

#if defined(__has_builtin)
#if __has_builtin(__builtin_amdgcn_s_wait_asynccnt)
#define WAIT_ASYNC0() __builtin_amdgcn_s_wait_asynccnt(0)
#else
#define WAIT_ASYNC0() asm volatile("s_wait_asynccnt 0x0" ::: "memory")
#endif
#else
#define WAIT_ASYNC0() asm volatile("s_wait_asynccnt 0x0" ::: "memory")
#endif

// ws layout (floats): qn[8192*256] | rn[8192*256] | rowSum[8192] | colSum[8192] | pos[1]

__global__ void zero_kernel(float* __restrict__ p, int n) {
    int i = blockIdx.x * blockDim.x + threadIdx.x;
    if (i < n) p[i] = 0.0f;
}

// One wave normalizes one 256-dim row; 8 rows per 256-thread block.
__global__ __launch_bounds__(256) void nrm_kernel(const float* __restrict__ vis,
                                                  const float* __restrict__ sent,
                                                  float* __restrict__ qn,
                                                  float* __restrict__ rn) {
    const int wave = threadIdx.x >> 5;
    const int lane = threadIdx.x & 31;
    int row = blockIdx.x * 8 + wave;            // 0..16383
    const float* src;
    float* dst;
    int r;
    if (row < B_TOTAL) { src = vis;  dst = qn; r = row; }
    else               { src = sent; dst = rn; r = row - B_TOTAL; }

    const float4* s4 = (const float4*)(src + (size_t)r * DIM);
    float4*       d4 = (float4*)(dst + (size_t)r * DIM);
    float4 x0 = s4[lane];
    float4 x1 = s4[lane + 32];
    float ss = x0.x * x0.x + x0.y * x0.y + x0.z * x0.z + x0.w * x0.w +
               x1.x * x1.x + x1.y * x1.y + x1.z * x1.z + x1.w * x1.w;
#pragma unroll
    for (int m = 1; m < 32; m <<= 1) ss += __shfl_xor(ss, m, 32);
    float nrm = sqrtf(ss);
    float sc  = 1.0f / fmaxf(nrm, 1e-12f);
    x0.x *= sc; x0.y *= sc; x0.z *= sc; x0.w *= sc;
    x1.x *= sc; x1.y *= sc; x1.z *= sc; x1.w *= sc;
    d4[lane]      = x0;
    d4[lane + 32] = x1;
}

// 128x128 sim tile per 256-thread block (8 waves); wave w owns a 16-row strip.
__global__ __launch_bounds__(256) void simtile_kernel(const float* __restrict__ qn,
                                                      const float* __restrict__ rn,
                                                      float* __restrict__ rowSum,
                                                      float* __restrict__ colSum,
                                                      float* __restrict__ posSum) {
    __shared__ __align__(16) float As[2][128 * LDST];
    __shared__ __align__(16) float Bs[2][128 * LDST];
    __shared__ float colAcc[128];

    const int t    = threadIdx.x;
    const int lane = t & 31;
    const int wave = t >> 5;
    const int ln   = lane & 15;       // N-within-subtile / M-within-strip
    const int hi   = lane >> 4;       // half-wave select (WMMA VGPR layout)
    const int rowStart = blockIdx.y * 128;
    const int colStart = blockIdx.x * 128;

    if (t < 128) colAcc[t] = 0.0f;

    // Stage a 128x32 chunk of A and B into LDS buffer `buf`.
    // Async path: direct global->LDS DMA, tracked with ASYNCcnt (no VGPR trip).
    auto stage = [&](int buf, int kc) {
#pragma unroll
        for (int j = 0; j < 4; ++j) {
            int idx = t + 256 * j;            // 0..1023 = 128 rows x 8 float4
            int r   = idx >> 3;
            int c4  = idx & 7;
            const float* ga = qn + (size_t)(rowStart + r) * DIM + kc + c4 * 4;
            const float* gb = rn + (size_t)(colStart + r) * DIM + kc + c4 * 4;
            float* la = &As[buf][r * LDST + c4 * 4];
            float* lb = &Bs[buf][r * LDST + c4 * 4];
#if USE_ASYNC
            __builtin_amdgcn_global_load_async_to_lds_b128(
                (AS1 v4i*)(uintptr_t)ga, (AS3 v4i*)(unsigned)(uintptr_t)la, 0, 0);
            __builtin_amdgcn_global_load_async_to_lds_b128(
                (AS1 v4i*)(uintptr_t)gb, (AS3 v4i*)(unsigned)(uintptr_t)lb, 0, 0);
#else
            if (kc + 32 < DIM) {              // prefetch next chunk (global_prefetch_b8)
                __builtin_prefetch(ga + 32, 0, 3);
                __builtin_prefetch(gb + 32, 0, 3);
            }
            *(float4*)la = *(const float4*)ga;
            *(float4*)lb = *(const float4*)gb;
#endif
        }
    };

    v8f acc[8];
#pragma unroll
    for (int i = 0; i < 8; ++i) acc[i] = (v8f){0, 0, 0, 0, 0, 0, 0, 0};

    stage(0, 0);                              // prologue: issue chunk 0

    for (int kc8 = 0; kc8 < DIM / 32; ++kc8) {
        const int buf = kc8 & 1;
#if USE_ASYNC
        WAIT_ASYNC0();                        // my async writes for `buf` landed
#endif
        __syncthreads();                      // all waves staged & done reading buf^1
        if (kc8 + 1 < DIM / 32)
            stage(buf ^ 1, (kc8 + 1) * 32);   // overlap next chunk's DMA with WMMAs

        const float* Abase = &As[buf][(wave * 16 + ln) * LDST + 2 * hi];
        const float* Bbase = &Bs[buf][ln * LDST + 2 * hi];
#pragma unroll
        for (int k = 0; k < 32; k += 4) {
            v2f a = *(const v2f*)(Abase + k); // 16x4 f32 A frag (K pair per half)
            v2f b[8];
#pragma unroll
            for (int nSub = 0; nSub < 8; ++nSub)
                b[nSub] = *(const v2f*)(Bbase + nSub * 16 * LDST + k);
#pragma unroll
            for (int nSub = 0; nSub < 8; ++nSub)
                acc[nSub] = __builtin_amdgcn_wmma_f32_16x16x4_f32(
                    false, a, false, b[nSub], (short)0, acc[nSub], false, false);
        }
    }

    // ---- fused epilogue: diag mask, partner extraction, exp, reductions ----
    float rowLocal[8];
#pragma unroll
    for (int v = 0; v < 8; ++v) rowLocal[v] = 0.0f;
    float posLocal = 0.0f;

#pragma unroll
    for (int nSub = 0; nSub < 8; ++nSub) {
        const int n = colStart + nSub * 16 + ln;
        float colLocal = 0.0f;
#pragma unroll
        for (int v = 0; v < 8; ++v) {
            const int m = rowStart + wave * 16 + v + 8 * hi;
            float s = acc[nSub][v] * INV_T;
            float e = (m == n) ? 0.0f : __expf(s);
            if (n == ((m + HALF_B) & (B_TOTAL - 1))) posLocal += s;
            rowLocal[v] += e;
            colLocal    += e;
        }
        colLocal += __shfl_xor(colLocal, 16, 32);   // merge half-waves (same col)
        if (hi == 0) atomicAdd(&colAcc[nSub * 16 + ln], colLocal);
    }

#pragma unroll
    for (int v = 0; v < 8; ++v) {                   // rows: reduce 16 lanes/half
        float rv = rowLocal[v];
        rv += __shfl_xor(rv, 1, 32);
        rv += __shfl_xor(rv, 2, 32);
        rv += __shfl_xor(rv, 4, 32);
        rv += __shfl_xor(rv, 8, 32);
        if (ln == 0)
            atomicAdd(&rowSum[rowStart + wave * 16 + v + 8 * hi], rv);
    }

    posLocal += __shfl_xor(posLocal, 1, 32);
    posLocal += __shfl_xor(posLocal, 2, 32);
    posLocal += __shfl_xor(posLocal, 4, 32);
    posLocal += __shfl_xor(posLocal, 8, 32);
    posLocal += __shfl_xor(posLocal, 16, 32);
    if (lane == 0) atomicAdd(posSum, posLocal);

    __syncthreads();
    if (t < 128) atomicAdd(&colSum[colStart + t], colAcc[t]);
}

__global__ __launch_bounds__(256) void finalize_kernel(const float* __restrict__ rowSum,
                                                       const float* __restrict__ colSum,
                                                       const float* __restrict__ posSum,
                                                       float* __restrict__ out) {
    __shared__ float red[256];
    float s = 0.0f;
    for (int i = threadIdx.x; i < B_TOTAL; i += 256)
        s += __logf(rowSum[i]) + __logf(colSum[i]);
    red[threadIdx.x] = s;
    __syncthreads();
#pragma unroll
    for (int off = 128; off > 0; off >>= 1) {
        if (threadIdx.x < off) red[threadIdx.x] += red[threadIdx.x + off];
        __syncthreads();
    }
    if (threadIdx.x == 0)
        out[0] = (red[0] - 2.0f * posSum[0]) / (float)B_TOTAL;
}

extern "C" void kernel_launch(void* const* d_in, const int* in_sizes, int n_in,
                              void* d_out, int out_size, void* d_ws, size_t ws_size,
                              hipStream_t stream) {
    const float* vis  = (const float*)d_in[0];
    const float* sent = (const float*)d_in[1];

    float* ws     = (float*)d_ws;
    float* qn     = ws;                              // 8192*256
    float* rn     = qn + (size_t)B_TOTAL * DIM;      // 8192*256
    float* rowSum = rn + (size_t)B_TOTAL * DIM;      // 8192
    float* colSum = rowSum + B_TOTAL;                // 8192
    float* pos    = colSum + B_TOTAL;                // 1

    zero_kernel<<<(2 * B_TOTAL + 1 + 255) / 256, 256, 0, stream>>>(rowSum, 2 * B_TOTAL + 1);

    nrm_kernel<<<(2 * B_TOTAL) / 8, 256, 0, stream>>>(vis, sent, qn, rn);

    dim3 grid(B_TOTAL / 128, B_TOTAL / 128);         // 64 x 64 tiles
    simtile_kernel<<<grid, 256, 0, stream>>>(qn, rn, rowSum, colSum, pos);

    finalize_kernel<<<1, 256, 0, stream>>>(rowSum, colSum, pos, (float*)d_out);
}